// LSTM_65326452572662
// MI455X (gfx1250) — compile-verified
//
#include <hip/hip_runtime.h>

// ---------------------------------------------------------------------------
// Fused LSTM controller for MI455X (gfx1250), fp32 end-to-end.
//   grid  = N/16 blocks, block = 32 threads (one wave32)
//   each wave owns 16 batch columns for all T steps (recurrence is serial in T)
// Gate GEMM per layer/step:  G[32 gates x 16 batch] = Wcat[32x16] * X[16x16]
// done with V_WMMA_F32_16X16X4_F32 chained over 4 K-slices, bias in initial C.
// Weight rows are PERMUTED so that each lane ends up holding i,f,g,o of 4
// hidden units for one batch column -> lane-local c/h update (no shuffles).
// ---------------------------------------------------------------------------

typedef float v2f __attribute__((ext_vector_type(2)));
typedef float v4f __attribute__((ext_vector_type(4)));
typedef float v8f __attribute__((ext_vector_type(8)));

// LDS ordering point within the single wave + compiler reorder fence.
#define DSBAR() asm volatile("s_wait_dscnt 0" ::: "memory")

__device__ __forceinline__ float fast_tanh(float x) {
#if __has_builtin(__builtin_amdgcn_tanhf)
    return __builtin_amdgcn_tanhf(x);                       // v_tanh_f32 (gfx1250 trans op)
#else
    // tanh(x) = 1 - 2/(exp2(2x*log2e)+1)   -> v_exp_f32 + v_rcp_f32
    float e = __builtin_amdgcn_exp2f(x * 2.8853900817779268f);
    return 1.0f - 2.0f * __builtin_amdgcn_rcpf(e + 1.0f);
#endif
}
__device__ __forceinline__ float fast_sig(float x) {
    return fmaf(0.5f, fast_tanh(0.5f * x), 0.5f);
}

__device__ __forceinline__ v8f wmma_k4(v2f a, v2f b, v8f c) {
    // D = A(16x4) * B(4x16) + C(16x16), fp32
    return __builtin_amdgcn_wmma_f32_16x16x4_f32(false, a, false, b, (short)0, c,
                                                 false, false);
}

// B operand (4x16 f32) lane layout assumed symmetric with documented A 16x4:
//   VGPR0: lanes 0-15 -> K=kBase+0, lanes 16-31 -> K=kBase+2 (col = lane&15)
//   VGPR1: K=kBase+1 / K=kBase+3
__device__ __forceinline__ v2f lds_bslice(const float* buf, int kBase, int col, int kOff) {
    v2f b;
    b.x = buf[(kBase + kOff + 0) * 16 + col];
    b.y = buf[(kBase + kOff + 1) * 16 + col];
    return b;
}

// permuted row p -> original LSTM gate row (gate*8 + unit), gate order i,f,g,o
// rows 0-7: units {0,1}; 16-23: units {2,3}; 8-15: units {4,5}; 24-31: units {6,7}
__device__ __forceinline__ int orig_row(int p) {
    int gate  = p & 3;
    int blk   = p >> 3;                       // 0..3
    int upair = ((blk & 1) << 1) | (blk >> 1); // 0,2,1,3 -> 0,2 swap
    int unit  = upair * 2 + ((p >> 2) & 1);
    return gate * 8 + unit;
}

__global__ __launch_bounds__(32)
void lstm_ctrl_kernel(
    const float* __restrict__ state, const float* __restrict__ dist,
    const float* __restrict__ e1W1, const float* __restrict__ e1b1,
    const float* __restrict__ e1W2, const float* __restrict__ e1b2,
    const float* __restrict__ e2W1, const float* __restrict__ e2b1,
    const float* __restrict__ e2W2, const float* __restrict__ e2b2,
    const float* __restrict__ Wih0, const float* __restrict__ Whh0,
    const float* __restrict__ bih0, const float* __restrict__ bhh0,
    const float* __restrict__ Wih1, const float* __restrict__ Whh1,
    const float* __restrict__ bih1, const float* __restrict__ bhh1,
    const float* __restrict__ decW, const float* __restrict__ decb,
    const float* __restrict__ finW, const float* __restrict__ finb,
    const float* __restrict__ fin2W, const float* __restrict__ fin2b,
    float* __restrict__ outp, int T, int Nn)
{
    const int lane = threadIdx.x;      // 0..31
    const int col  = lane & 15;        // batch column within tile / A row
    const int half = lane >> 4;        // 0 or 1
    const int kOff = half << 1;        // K sub-offset for A/B operands
    const int gcol = blockIdx.x * 16 + col;

    __shared__ float bufA[256];        // layer0 B staging: rows 8-15 = h0 state
    __shared__ float bufB[256];        // layer1 B staging: rows 0-7 = h_l0, 8-15 = h1
    __shared__ float tmpS[256];        // init-time c distribution

    // ---- load permuted A operands + fused biases into registers ------------
    const float* WihL[2] = { Wih0, Wih1 };
    const float* WhhL[2] = { Whh0, Whh1 };
    const float* bihL[2] = { bih0, bih1 };
    const float* bhhL[2] = { bhh0, bhh1 };

    v2f Aw[2][2][4];                   // [layer][gate-tile][k-slice]
    v8f BiasT[2][2];                   // [layer][gate-tile] (broadcast over cols)
#pragma unroll
    for (int l = 0; l < 2; ++l) {
#pragma unroll
        for (int tI = 0; tI < 2; ++tI) {
            const int orow = orig_row(tI * 16 + col);   // A row = lane&15
#pragma unroll
            for (int ks = 0; ks < 4; ++ks) {
                v2f a;
#pragma unroll
                for (int cc = 0; cc < 2; ++cc) {
                    int k = ks * 4 + kOff + cc;         // concat feature: x|h
                    float w = (k < 8) ? WihL[l][orow * 8 + k]
                                      : WhhL[l][orow * 8 + (k - 8)];
                    if (cc == 0) a.x = w; else a.y = w;
                }
                Aw[l][tI][ks] = a;
            }
            v8f bv;
#pragma unroll
            for (int r = 0; r < 8; ++r) {
                int br = orig_row(tI * 16 + r + half * 8);  // D row for this lane
                bv[r] = bihL[l][br] + bhhL[l][br];
            }
            BiasT[l][tI] = bv;
        }
    }

    // ---- decoder weights in registers --------------------------------------
    float dWr[4][8], dbr[4], hWr[4][4], hbr[4];
    {
        const float* fw = half ? fin2W : finW;   // lane<16 -> mu, lane>=16 -> sigma_sq
        const float* fb = half ? fin2b : finb;
#pragma unroll
        for (int d = 0; d < 4; ++d) {
            dbr[d] = decb[d];
#pragma unroll
            for (int u = 0; u < 8; ++u) dWr[d][u] = decW[d * 8 + u];
        }
#pragma unroll
        for (int a = 0; a < 4; ++a) {
            hbr[a] = fb[a];
#pragma unroll
            for (int d = 0; d < 4; ++d) hWr[a][d] = fw[a * 4 + d];
        }
    }

    // ---- encoders: lanes 0-15 -> h (enc1), lanes 16-31 -> c (enc2) ----------
    {
        const float* W1 = half ? e2W1 : e1W1;
        const float* b1 = half ? e2b1 : e1b1;
        const float* W2 = half ? e2W2 : e1W2;
        const float* b2 = half ? e2b2 : e1b2;
        float sv[16];
#pragma unroll
        for (int i = 0; i < 16; ++i) sv[i] = state[gcol * 16 + i];
        float eh[4];
#pragma unroll
        for (int j = 0; j < 4; ++j) {
            float acc = b1[j];
#pragma unroll
            for (int i = 0; i < 16; ++i) acc = fmaf(W1[j * 16 + i], sv[i], acc);
            eh[j] = fmaxf(acc, 0.0f);
        }
        float ee[16];
#pragma unroll
        for (int u = 0; u < 16; ++u) {
            float acc = b2[u];
#pragma unroll
            for (int j = 0; j < 4; ++j) acc = fmaf(W2[u * 4 + j], eh[j], acc);
            ee[u] = fmaxf(acc, 0.0f);
        }
        if (half == 0) {                       // h0: layer0 -> bufA, layer1 -> bufB
#pragma unroll
            for (int u = 0; u < 8; ++u) bufA[(8 + u) * 16 + col] = ee[u];
#pragma unroll
            for (int u = 0; u < 8; ++u) bufB[(8 + u) * 16 + col] = ee[8 + u];
        } else {                               // c0 -> tmp for redistribution
#pragma unroll
            for (int u = 0; u < 16; ++u) tmpS[u * 16 + col] = ee[u];
        }
    }
    DSBAR();

    float c0s[4], c1s[4];                      // lane-local cell state (4 units)
#pragma unroll
    for (int ul = 0; ul < 4; ++ul) {
        int gu = half * 4 + ul;
        c0s[ul] = tmpS[gu * 16 + col];
        c1s[ul] = tmpS[(8 + gu) * 16 + col];
    }

    // gate extraction from the two D tiles (lane-local after row permutation)
#define GATE(G0, G1, ul, idx) (((ul) < 2) ? (G0)[(ul) * 4 + (idx)] : (G1)[((ul) - 2) * 4 + (idx)])

    const size_t muBase  = 0;
    const size_t sgBase  = (size_t)T * (size_t)Nn * 4;

    for (int t = 0; t < T; ++t) {
        // ---- x_t gathered straight into B lane layout (b64 per slice) ------
        const float* xp = dist + ((size_t)t * Nn + gcol) * 8;
        v2f Bx0, Bx1;
        Bx0.x = xp[kOff + 0]; Bx0.y = xp[kOff + 1];       // K = 0..3
        Bx1.x = xp[4 + kOff + 0]; Bx1.y = xp[4 + kOff + 1]; // K = 4..7
        if (t + 1 < T) __builtin_prefetch(xp + (size_t)Nn * 8, 0, 1);

        // ---- layer 0 gates --------------------------------------------------
        v2f Bh0 = lds_bslice(bufA, 8,  col, kOff);        // K = 8..11  (h0 state)
        v2f Bh1 = lds_bslice(bufA, 12, col, kOff);        // K = 12..15
        v8f g0 = BiasT[0][0], g1 = BiasT[0][1];
        g0 = wmma_k4(Aw[0][0][0], Bx0, g0);  g1 = wmma_k4(Aw[0][1][0], Bx0, g1);
        g0 = wmma_k4(Aw[0][0][1], Bx1, g0);  g1 = wmma_k4(Aw[0][1][1], Bx1, g1);
        g0 = wmma_k4(Aw[0][0][2], Bh0, g0);  g1 = wmma_k4(Aw[0][1][2], Bh0, g1);
        g0 = wmma_k4(Aw[0][0][3], Bh1, g0);  g1 = wmma_k4(Aw[0][1][3], Bh1, g1);

        float h0n[4];
#pragma unroll
        for (int ul = 0; ul < 4; ++ul) {
            float gi = GATE(g0, g1, ul, 0), gf = GATE(g0, g1, ul, 1);
            float gg = GATE(g0, g1, ul, 2), go = GATE(g0, g1, ul, 3);
            float cn = fmaf(fast_sig(gf), c0s[ul], fast_sig(gi) * fast_tanh(gg));
            c0s[ul] = cn;
            h0n[ul] = fast_sig(go) * fast_tanh(cn);
        }
#pragma unroll
        for (int ul = 0; ul < 4; ++ul) {
            int gu = half * 4 + ul;
            bufA[(8 + gu) * 16 + col] = h0n[ul];   // h0 state for t+1
            bufB[gu * 16 + col]       = h0n[ul];   // layer1 input features
        }
        DSBAR();

        // ---- layer 1 gates --------------------------------------------------
        v2f B0 = lds_bslice(bufB, 0,  col, kOff);
        v2f B1 = lds_bslice(bufB, 4,  col, kOff);
        v2f B2 = lds_bslice(bufB, 8,  col, kOff);         // h1 state (pre-update)
        v2f B3 = lds_bslice(bufB, 12, col, kOff);
        v8f G0 = BiasT[1][0], G1 = BiasT[1][1];
        G0 = wmma_k4(Aw[1][0][0], B0, G0);  G1 = wmma_k4(Aw[1][1][0], B0, G1);
        G0 = wmma_k4(Aw[1][0][1], B1, G0);  G1 = wmma_k4(Aw[1][1][1], B1, G1);
        G0 = wmma_k4(Aw[1][0][2], B2, G0);  G1 = wmma_k4(Aw[1][1][2], B2, G1);
        G0 = wmma_k4(Aw[1][0][3], B3, G0);  G1 = wmma_k4(Aw[1][1][3], B3, G1);

        float h1n[4];
#pragma unroll
        for (int ul = 0; ul < 4; ++ul) {
            float gi = GATE(G0, G1, ul, 0), gf = GATE(G0, G1, ul, 1);
            float gg = GATE(G0, G1, ul, 2), go = GATE(G0, G1, ul, 3);
            float cn = fmaf(fast_sig(gf), c1s[ul], fast_sig(gi) * fast_tanh(gg));
            c1s[ul] = cn;
            h1n[ul] = fast_sig(go) * fast_tanh(cn);
        }
#pragma unroll
        for (int ul = 0; ul < 4; ++ul) {
            int gu = half * 4 + ul;
            bufB[(8 + gu) * 16 + col] = h1n[ul];   // h1 state for t+1 + decoder in
        }
        DSBAR();

        // ---- decoder + heads (lane<16: mu, lane>=16: sigma_sq) --------------
        float hv[8];
#pragma unroll
        for (int u = 0; u < 8; ++u) hv[u] = bufB[(8 + u) * 16 + col];
        float dec[4];
#pragma unroll
        for (int d = 0; d < 4; ++d) {
            float acc = dbr[d];
#pragma unroll
            for (int u = 0; u < 8; ++u) acc = fmaf(dWr[d][u], hv[u], acc);
            dec[d] = fmaxf(acc, 0.0f);
        }
        v4f res;
#pragma unroll
        for (int a = 0; a < 4; ++a) {
            float acc = hbr[a];
#pragma unroll
            for (int d = 0; d < 4; ++d) acc = fmaf(hWr[a][d], dec[d], acc);
            res[a] = acc;
        }
        size_t off = (half ? sgBase : muBase) + ((size_t)t * Nn + gcol) * 4;
        *(v4f*)(outp + off) = res;                 // global_store_b128, 16B aligned
    }
#undef GATE
}

extern "C" void kernel_launch(void* const* d_in, const int* in_sizes, int n_in,
                              void* d_out, int out_size, void* d_ws, size_t ws_size,
                              hipStream_t stream) {
    (void)n_in; (void)out_size; (void)d_ws; (void)ws_size;
    const float* state = (const float*)d_in[0];
    const float* dist  = (const float*)d_in[1];
    const float* e1W1  = (const float*)d_in[2];
    const float* e1b1  = (const float*)d_in[3];
    const float* e1W2  = (const float*)d_in[4];
    const float* e1b2  = (const float*)d_in[5];
    const float* e2W1  = (const float*)d_in[6];
    const float* e2b1  = (const float*)d_in[7];
    const float* e2W2  = (const float*)d_in[8];
    const float* e2b2  = (const float*)d_in[9];
    const float* Wih0  = (const float*)d_in[10];
    const float* Whh0  = (const float*)d_in[11];
    const float* bih0  = (const float*)d_in[12];
    const float* bhh0  = (const float*)d_in[13];
    const float* Wih1  = (const float*)d_in[14];
    const float* Whh1  = (const float*)d_in[15];
    const float* bih1  = (const float*)d_in[16];
    const float* bhh1  = (const float*)d_in[17];
    const float* decW  = (const float*)d_in[18];
    const float* decb  = (const float*)d_in[19];
    const float* finW  = (const float*)d_in[20];
    const float* finb  = (const float*)d_in[21];
    const float* fin2W = (const float*)d_in[22];
    const float* fin2b = (const float*)d_in[23];

    const int Nn = in_sizes[0] / 16;          // batch (4096)
    const int T  = in_sizes[1] / (Nn * 8);    // seq len (1000)

    lstm_ctrl_kernel<<<dim3(Nn / 16), dim3(32), 0, stream>>>(
        state, dist,
        e1W1, e1b1, e1W2, e1b2, e2W1, e2b1, e2W2, e2b2,
        Wih0, Whh0, bih0, bhh0, Wih1, Whh1, bih1, bhh1,
        decW, decb, finW, finb, fin2W, fin2b,
        (float*)d_out, T, Nn);
}